// DNF_21912923144502
// MI455X (gfx1250) — compile-verified
//
#include <hip/hip_runtime.h>
#include <math.h>

// DNF semi-symbolic layers for MI455X (gfx1250, wave32, WMMA + async-to-LDS).
//
// layer(A[M,K], W[N,K]) -> out[M,N]:
//   lin    = A @ W^T                      (WMMA f32 16x16x4)
//   absacc = |A| @ |W|^T                  (WMMA f32 16x16x4)
//   amax   = max_k |A[m,k] * W[n,k]|      (VALU v_pk_mul + v_max3 |src|)
//   CONJ:  out = tanh(lin + 0.1*(amax - absacc))
//   else:  out =       lin + 0.1*(absacc - amax)
//
// Staging: double-buffered LDS; GLOBAL_LOAD_ASYNC_TO_LDS_B128 (ASYNCcnt)
// prefetches chunk k+1 while chunk k is computed.

typedef float v2f __attribute__((ext_vector_type(2)));
typedef float v8f __attribute__((ext_vector_type(8)));
typedef int   vi4 __attribute__((vector_size(4 * sizeof(int))));

#define DELTA_F 0.1f

#define AS_GLOBAL __attribute__((address_space(1)))
#define AS_LDS    __attribute__((address_space(3)))

#if __has_builtin(__builtin_amdgcn_global_load_async_to_lds_b128)
#define HAVE_ASYNC_LDS 1
#else
#define HAVE_ASYNC_LDS 0
#endif

__device__ __forceinline__ void copy16_to_lds(float* lds_dst, const float* gsrc) {
#if HAVE_ASYNC_LDS
    // Async DMA: global -> LDS, no VGPR round-trip, tracked by ASYNCcnt.
    __builtin_amdgcn_global_load_async_to_lds_b128(
        (AS_GLOBAL vi4*)gsrc, (AS_LDS vi4*)lds_dst, 0, 0);
#else
    *(float4*)lds_dst = *(const float4*)gsrc;
#endif
}

__device__ __forceinline__ void wait_async_lds() {
#if HAVE_ASYNC_LDS
#if __has_builtin(__builtin_amdgcn_s_wait_asynccnt)
    __builtin_amdgcn_s_wait_asynccnt(0);
#else
    asm volatile("s_wait_asynccnt 0" ::: "memory");
#endif
#endif
}

template <int N, int K, bool CONJ>
__global__ __launch_bounds__(128) void dnf_layer(const float* __restrict__ A,
                                                 const float* __restrict__ W,
                                                 float* __restrict__ out) {
    constexpr int BM = 32;   // batch tile
    constexpr int BN = 32;   // output tile
    constexpr int KC = 64;   // K chunk staged in LDS
    constexpr int P  = 68;   // LDS row pitch (floats): 16B-aligned rows,
                             // n*68 mod 64 -> 16 distinct banks (conflict-free)

    __shared__ __align__(16) float xs[2][BM * P];   // A tiles (double buffer)
    __shared__ __align__(16) float wt[2][BN * P];   // W tiles (double buffer)

    const int tid  = threadIdx.x;
    const int lane = tid & 31;
    const int wave = tid >> 5;
    const int wm   = (wave >> 1) << 4;   // wave's 16-row sub-tile
    const int wn   = (wave & 1) << 4;    // wave's 16-col sub-tile
    const int lr   = lane & 15;          // row/col within fragment
    const int lh   = lane >> 4;          // lane half (K/M split per ISA layout)

    const int m0 = blockIdx.x * BM;
    const int n0 = blockIdx.y * BN;

    // Cooperative staging of one KC-wide chunk into LDS buffer `buf`.
    auto stage = [&](int buf, int k0) {
#pragma unroll
        for (int l = tid; l < BM * (KC / 4); l += 128) {
            const int r = l / (KC / 4);
            const int c = (l % (KC / 4)) * 4;
            copy16_to_lds(&xs[buf][r * P + c],
                          &A[(size_t)(m0 + r) * K + k0 + c]);
        }
#pragma unroll
        for (int l = tid; l < BN * (KC / 4); l += 128) {
            const int r = l / (KC / 4);
            const int c = (l % (KC / 4)) * 4;
            copy16_to_lds(&wt[buf][r * P + c],
                          &W[(size_t)(n0 + r) * K + k0 + c]);
        }
    };

    v8f acc_lin = {};
    v8f acc_abs = {};
    float amax[8];
#pragma unroll
    for (int i = 0; i < 8; ++i) amax[i] = 0.0f;  // products are >= 0

    // Prologue: fill buffer 0.
    stage(0, 0);
    wait_async_lds();
    __syncthreads();

    for (int k0 = 0; k0 < K; k0 += KC) {
        const int cb = (k0 / KC) & 1;

        // Prefetch next chunk into the other buffer while we compute.
        if (k0 + KC < K) stage(cb ^ 1, k0 + KC);

        const float* xrow = &xs[cb][(wm + lr) * P];  // A frag row (M = lr)
        const float* wrow = &wt[cb][(wn + lr) * P];  // B^T frag row (N = lr)

        // ---- WMMA: lin and abs dot products, K in steps of 4 ----
        // 16x4 A frag / 4x16 B frag: lane L holds Mat[L%16][k + 2*(L/16) + v]
#pragma unroll
        for (int kk = 0; kk < KC; kk += 4) {
            v2f a = *(const v2f*)(xrow + kk + 2 * lh);
            v2f b = *(const v2f*)(wrow + kk + 2 * lh);
            acc_lin = __builtin_amdgcn_wmma_f32_16x16x4_f32(
                false, a, false, b, (short)0, acc_lin, false, false);
            v2f aa, bb;
            aa.x = __builtin_fabsf(a.x); aa.y = __builtin_fabsf(a.y);
            bb.x = __builtin_fabsf(b.x); bb.y = __builtin_fabsf(b.y);
            acc_abs = __builtin_amdgcn_wmma_f32_16x16x4_f32(
                false, aa, false, bb, (short)0, acc_abs, false, false);
        }

        // ---- max_k |a*w| : accumulator layout matches C fragment ----
        // lane L, reg i  <->  (M = wm + i + 8*lh, N = wn + lr)
        // Batch all 8 row loads per k-quad so ds_load_b128s stay in flight
        // (avoids one s_wait_dscnt per load in the hot loop).
#pragma unroll 2
        for (int k = 0; k < KC; k += 4) {
            const float4 wv = *(const float4*)(wrow + k);
            float4 xv[8];
#pragma unroll
            for (int i = 0; i < 8; ++i)
                xv[i] = *(const float4*)&xs[cb][(wm + i + 8 * lh) * P + k];
#pragma unroll
            for (int i = 0; i < 8; ++i) {
                amax[i] = fmaxf(amax[i], __builtin_fabsf(xv[i].x * wv.x));
                amax[i] = fmaxf(amax[i], __builtin_fabsf(xv[i].y * wv.y));
                amax[i] = fmaxf(amax[i], __builtin_fabsf(xv[i].z * wv.z));
                amax[i] = fmaxf(amax[i], __builtin_fabsf(xv[i].w * wv.w));
            }
        }

        // Next chunk's async loads must land before anyone reads buffer cb^1.
        wait_async_lds();
        __syncthreads();
    }

    // ---- combine + store (64B-contiguous per half-wave per reg) ----
#pragma unroll
    for (int i = 0; i < 8; ++i) {
        const float bias = CONJ ? (amax[i] - acc_abs[i])
                                : (acc_abs[i] - amax[i]);
        float v = acc_lin[i] + DELTA_F * bias;
        if (CONJ) v = tanhf(v);
        const int m = m0 + wm + i + 8 * lh;
        const int n = n0 + wn + lr;
        out[(size_t)m * N + n] = v;
    }
}

extern "C" void kernel_launch(void* const* d_in, const int* in_sizes, int n_in,
                              void* d_out, int out_size, void* d_ws, size_t ws_size,
                              hipStream_t stream) {
    (void)in_sizes; (void)n_in; (void)out_size; (void)ws_size;

    const float* x  = (const float*)d_in[0];   // (1024, 512)
    const float* Wc = (const float*)d_in[1];   // (512, 512)
    const float* Wd = (const float*)d_in[2];   // (128, 512)
    float* out  = (float*)d_out;               // (1024, 128)
    float* conj = (float*)d_ws;                // (1024, 512) intermediate, 2 MB

    // Layer 1: conj = tanh(semi_symbolic(x, Wc, conjunction=true))
    dnf_layer<512, 512, true>
        <<<dim3(1024 / 32, 512 / 32), 128, 0, stream>>>(x, Wc, conj);

    // Layer 2: out = semi_symbolic(conj, Wd, conjunction=false)
    dnf_layer<128, 512, false>
        <<<dim3(1024 / 32, 128 / 32), 128, 0, stream>>>(conj, Wd, out);
}